// NTMMemory_43301860278666
// MI455X (gfx1250) — compile-verified
//
#include <hip/hip_runtime.h>

#define B_ 32
#define H_ 4
#define N_ 8192
#define D_ 64
#define EPSF 1e-10f
#define TS 68   // padded LDS tile row stride (floats) to dodge bank conflicts

typedef float v2f __attribute__((ext_vector_type(2)));
typedef float v8f __attribute__((ext_vector_type(8)));

// workspace layout (in floats)
#define OFF_E   0
#define OFF_W   (B_*H_*N_)                 // 1,048,576
#define OFF_KN  (2*B_*H_*N_)               // 2,097,152
#define OFF_S1  (OFF_KN + B_*H_*D_)
#define OFF_S2  (OFF_S1 + B_*H_)

// ---------------------------------------------------------------- zero init
__global__ void ntm_zero(float* __restrict__ S1, float* __restrict__ S2,
                         float* __restrict__ out) {
  int i = blockIdx.x * 256 + threadIdx.x;
  if (i < B_*H_) { S1[i] = 0.f; S2[i] = 0.f; }
  if (i < B_*H_*D_) out[i] = 0.f;
}

// ---------------------------------------------------------------- normalize k
__global__ void ntm_knorm(const float* __restrict__ k, float* __restrict__ KN) {
  int t = threadIdx.x;            // t = b*H + h, 128 threads
  if (t >= B_*H_) return;
  const float* kp = k + t * D_;
  float ss = 0.f;
  #pragma unroll
  for (int d = 0; d < D_; ++d) ss = fmaf(kp[d], kp[d], ss);
  float inv = 1.f / (sqrtf(ss) + EPSF);
  #pragma unroll
  for (int d = 0; d < D_; ++d) KN[t * D_ + d] = kp[d] * inv;
}

// ------------------------------------------- pass 1: content addressing (WMMA)
// grid (B, 16), block 128 (4 waves). Each block: 512 rows in 8 chunks of 64.
__global__ __launch_bounds__(128) void ntm_content(
    const float* __restrict__ mem, const float* __restrict__ beta,
    const float* __restrict__ kn, float* __restrict__ E, float* __restrict__ S1)
{
  __shared__ float tile[4 * 16 * TS];
  __shared__ float kknp[16 * D_];          // zero-padded to 16 WMMA A rows
  const int b = blockIdx.x, by = blockIdx.y;
  const int tid = threadIdx.x, lane = tid & 31, wave = tid >> 5;

  for (int i = tid; i < 16 * D_; i += 128)
    kknp[i] = (i < H_ * D_) ? kn[b*H_*D_ + i] : 0.f;
  float bet[H_];
  #pragma unroll
  for (int r = 0; r < H_; ++r) bet[r] = beta[b*H_ + r];
  float esum[H_] = {0.f, 0.f, 0.f, 0.f};

  const int hA   = lane & 15;              // A-matrix row (head, zero-padded)
  const int koff = (lane >> 4) * 2;        // lanes 16-31 hold K=2,3
  __syncthreads();

  // A-fragments are kernel-invariant: hoist all 16 into registers once.
  v2f afr[16];
  #pragma unroll
  for (int d4 = 0; d4 < 16; ++d4)
    afr[d4] = *(const v2f*)&kknp[hA * D_ + d4 * 4 + koff];

  for (int c = 0; c < 8; ++c) {
    const int n0 = by * 512 + c * 64;
    __syncthreads();
    // cooperative coalesced load + prefetch of next chunk
    for (int q = tid; q < 1024; q += 128) {
      int row = q >> 4, cv = q & 15;
      float4 v = ((const float4*)(mem + ((size_t)b*N_ + n0 + row) * D_))[cv];
      *(float4*)&tile[(row >> 4)*16*TS + (row & 15)*TS + cv*4] = v;
      if (c < 7)
        __builtin_prefetch(mem + ((size_t)b*N_ + n0 + 64 + row) * D_ + cv*4, 0, 1);
    }
    __syncthreads();

    float* mt = &tile[wave * 16 * TS];
    // row norms: two lanes per row, combine with shfl_xor(16)
    const int rrow = lane & 15, half = lane >> 4;
    float ss = 0.f;
    const float* rp = &mt[rrow * TS + half * 32];
    #pragma unroll
    for (int i = 0; i < 32; ++i) { float v = rp[i]; ss = fmaf(v, v, ss); }
    ss += __shfl_xor(ss, 16);
    const float rn = 1.f / (sqrtf(ss) + EPSF);

    // batch all B-fragments, then run the WMMA chain
    v2f bfr[16];
    #pragma unroll
    for (int d4 = 0; d4 < 16; ++d4)
      bfr[d4] = *(const v2f*)&mt[rrow * TS + d4 * 4 + koff];

    v8f acc = {0.f,0.f,0.f,0.f,0.f,0.f,0.f,0.f};
    #pragma unroll
    for (int d4 = 0; d4 < 16; ++d4)
      acc = __builtin_amdgcn_wmma_f32_16x16x4_f32(
          false, afr[d4], false, bfr[d4], (short)0, acc, false, false);

    if (lane < 16) {
      const int n = n0 + wave * 16 + lane;
      #pragma unroll
      for (int r = 0; r < H_; ++r) {
        // softmax w/o max-sub: beta*cos bounded in (-5,5)
        float e = __expf(bet[r] * acc[r] * rn);
        E[((size_t)(b*H_ + r)) * N_ + n] = e;
        esum[r] += e;
      }
    }
  }
  #pragma unroll
  for (int r = 0; r < H_; ++r) {
    float v = esum[r];
    #pragma unroll
    for (int off = 16; off; off >>= 1) v += __shfl_xor(v, off);
    if (lane == 0) atomicAdd(&S1[b*H_ + r], v);
  }
}

// ------------------------- pass 2: gate + circular shift + sharpen (elementwise)
__global__ __launch_bounds__(256) void ntm_shift(
    const float* __restrict__ E, const float* __restrict__ S1,
    const float* __restrict__ pw, const float* __restrict__ g,
    const float* __restrict__ s, const float* __restrict__ gamma,
    float* __restrict__ Wbuf, float* __restrict__ S2)
{
  const int bh = blockIdx.x;
  const int n0 = blockIdx.y * 1024;
  const float s1inv = 1.f / S1[bh];
  const float gv = g[bh], om = 1.f - gv;
  const float sh0 = s[bh*3+0], sh1 = s[bh*3+1], sh2 = s[bh*3+2];
  const float gam = gamma[bh];
  const float* Eb = E  + (size_t)bh * N_;
  const float* pb = pw + (size_t)bh * N_;
  float*       Wb = Wbuf + (size_t)bh * N_;
  float lsum = 0.f;
  for (int i = threadIdx.x; i < 1024; i += 256) {
    int n  = n0 + i;
    int nm = (n + N_ - 1) & (N_ - 1);
    int np = (n + 1) & (N_ - 1);
    float wgp = gv*Eb[np]*s1inv + om*pb[np];
    float wgc = gv*Eb[n ]*s1inv + om*pb[n ];
    float wgm = gv*Eb[nm]*s1inv + om*pb[nm];
    float ws  = sh0*wgp + sh1*wgc + sh2*wgm;   // ws[i]=s0*wg[i+1]+s1*wg[i]+s2*wg[i-1]
    float wgam = (ws > 0.f) ? __expf(gam * __logf(ws)) : 0.f;
    Wb[n] = wgam;
    lsum += wgam;
  }
  __shared__ float red[256];
  red[threadIdx.x] = lsum; __syncthreads();
  for (int st = 128; st; st >>= 1) {
    if (threadIdx.x < st) red[threadIdx.x] += red[threadIdx.x + st];
    __syncthreads();
  }
  if (threadIdx.x == 0) atomicAdd(&S2[bh], red[0]);
}

// ---------------- pass 3: erase/add update in LDS + WMMA read-reduction
// grid (B, 16), block 128. C accumulators persist across 8 chunks; 2nd memory
// pass should be L2-resident (64 MB < 192 MB L2).
__global__ __launch_bounds__(128) void ntm_writeread(
    const float* __restrict__ mem, const float* __restrict__ Wbuf,
    const float* __restrict__ S2, const float* __restrict__ erase_,
    const float* __restrict__ add_, float* __restrict__ out)
{
  __shared__ float tile[4 * 16 * TS];
  __shared__ float wls[4 * 16 * 16];       // [wave][head(16, zero-padded)][row]
  __shared__ float els[H_ * D_], eadd[H_ * D_];
  const int b = blockIdx.x, by = blockIdx.y;
  const int tid = threadIdx.x, lane = tid & 31, wave = tid >> 5;

  for (int i = tid; i < H_*D_; i += 128) {
    els[i]  = erase_[b*H_*D_ + i];
    eadd[i] = add_[b*H_*D_ + i];
  }
  for (int i = tid; i < 4 * 16 * 16; i += 128) wls[i] = 0.f;  // pad rows stay 0

  const int hA   = lane & 15;
  const int koff = (lane >> 4) * 2;

  v8f accs[4];
  #pragma unroll
  for (int t = 0; t < 4; ++t) accs[t] = (v8f){0.f,0.f,0.f,0.f,0.f,0.f,0.f,0.f};

  for (int c = 0; c < 8; ++c) {
    const int n0 = by * 512 + c * 64;
    __syncthreads();
    for (int q = tid; q < 1024; q += 128) {          // stage 64 rows
      int row = q >> 4, cv = q & 15;
      float4 v = ((const float4*)(mem + ((size_t)b*N_ + n0 + row) * D_))[cv];
      *(float4*)&tile[(row >> 4)*16*TS + (row & 15)*TS + cv*4] = v;
      if (c < 7)
        __builtin_prefetch(mem + ((size_t)b*N_ + n0 + 64 + row) * D_ + cv*4, 0, 1);
    }
    for (int q = tid; q < 4*H_*16; q += 128) {       // normalized w for 64 rows
      int wv = q >> 6, h = (q >> 4) & 3, r = q & 15;
      wls[wv*256 + h*16 + r] =
          Wbuf[((size_t)(b*H_ + h)) * N_ + n0 + wv*16 + r] / S2[b*H_ + h];
    }
    __syncthreads();

    {   // sequential 4-head erase/add update, in LDS
      float* mt = &tile[wave * 16 * TS];
      const float* wl = &wls[wave * 256];
      int row = lane >> 1, coff = (lane & 1) * 32;
      float w0 = wl[0*16+row], w1 = wl[1*16+row], w2 = wl[2*16+row], w3 = wl[3*16+row];
      float* rp = &mt[row * TS + coff];
      #pragma unroll
      for (int i = 0; i < 32; ++i) {
        int col = coff + i;
        float m = rp[i];
        m = m * (1.f - w0*els[0*D_+col]) + w0*eadd[0*D_+col];
        m = m * (1.f - w1*els[1*D_+col]) + w1*eadd[1*D_+col];
        m = m * (1.f - w2*els[2*D_+col]) + w2*eadd[2*D_+col];
        m = m * (1.f - w3*els[3*D_+col]) + w3*eadd[3*D_+col];
        rp[i] = m;
      }
    }
    __syncthreads();

    // reading[h, d] += sum_n w[h,n] * mem'[n,d]  via WMMA, 4 D-tiles of 16.
    // Batch all fragments first, then 4 independent WMMA chains.
    const float* mt = &tile[wave * 16 * TS];
    const float* wl = &wls[wave * 256];
    const int col = lane & 15;

    v2f awf[4];
    #pragma unroll
    for (int nk4 = 0; nk4 < 4; ++nk4)
      awf[nk4] = *(const v2f*)&wl[hA * 16 + nk4 * 4 + koff];

    v2f bfr[16];
    #pragma unroll
    for (int nk4 = 0; nk4 < 4; ++nk4)
      #pragma unroll
      for (int t = 0; t < 4; ++t) {
        bfr[nk4*4+t][0] = mt[(nk4*4 + koff)     * TS + t*16 + col];
        bfr[nk4*4+t][1] = mt[(nk4*4 + koff + 1) * TS + t*16 + col];
      }

    #pragma unroll
    for (int nk4 = 0; nk4 < 4; ++nk4)
      #pragma unroll
      for (int t = 0; t < 4; ++t)
        accs[t] = __builtin_amdgcn_wmma_f32_16x16x4_f32(
            false, awf[nk4], false, bfr[nk4*4+t], (short)0, accs[t], false, false);
  }

  if (lane < 16) {
    #pragma unroll
    for (int t = 0; t < 4; ++t)
      #pragma unroll
      for (int r = 0; r < H_; ++r)
        atomicAdd(&out[(b*H_ + r) * D_ + t*16 + lane], accs[t][r]);
  }
}

// ---------------------------------------------------------------- launcher
extern "C" void kernel_launch(void* const* d_in, const int* in_sizes, int n_in,
                              void* d_out, int out_size, void* d_ws, size_t ws_size,
                              hipStream_t stream) {
  const float* mem   = (const float*)d_in[0];
  const float* k     = (const float*)d_in[1];
  const float* beta  = (const float*)d_in[2];
  const float* pw    = (const float*)d_in[3];
  const float* g     = (const float*)d_in[4];
  const float* s     = (const float*)d_in[5];
  const float* gamma = (const float*)d_in[6];
  const float* erase = (const float*)d_in[7];
  const float* add   = (const float*)d_in[8];
  float* out = (float*)d_out;
  float* ws  = (float*)d_ws;
  float* E  = ws + OFF_E;
  float* Wb = ws + OFF_W;
  float* KN = ws + OFF_KN;
  float* S1 = ws + OFF_S1;
  float* S2 = ws + OFF_S2;

  ntm_zero<<<dim3((B_*H_*D_ + 255) / 256), 256, 0, stream>>>(S1, S2, out);
  ntm_knorm<<<1, 128, 0, stream>>>(k, KN);
  ntm_content<<<dim3(B_, 16), 128, 0, stream>>>(mem, beta, KN, E, S1);
  ntm_shift<<<dim3(B_*H_, N_/1024), 256, 0, stream>>>(E, S1, pw, g, s, gamma, Wb, S2);
  ntm_writeread<<<dim3(B_, 16), 128, 0, stream>>>(mem, Wb, S2, erase, add, out);
}